// RnCLoss_29669634080861
// MI455X (gfx1250) — compile-verified
//
#include <hip/hip_runtime.h>
#include <math.h>

typedef __attribute__((ext_vector_type(2))) float v2f;
typedef __attribute__((ext_vector_type(4))) float v4f;
typedef __attribute__((ext_vector_type(8))) float v8f;

#define N     4096
#define D     64
#define TILES (N / 16)        // 256 tiles per dimension
#define BLK   256
#define KPT   (N / BLK)       // 16 k-values per thread

// ---------------------------------------------------------------- init ----
__global__ void rnc_init_kernel(double* acc) {
    acc[0] = 0.0;
}

// ------------------------------------------------------- row squared norms
__global__ void rnc_sq_kernel(const float* __restrict__ F,
                              float* __restrict__ sq) {
    int i = blockIdx.x * blockDim.x + threadIdx.x;
    if (i < N) {
        const v4f* row = (const v4f*)(F + (size_t)i * D);
        float s = 0.f;
#pragma unroll
        for (int k = 0; k < D / 4; ++k) {
            v4f v = row[k];
            s += v.x * v.x + v.y * v.y + v.z * v.z + v.w * v.w;
        }
        sq[i] = s;
    }
}

// ------------------------------------------ WMMA fp32 Gram -> logits matrix
// One wave computes one 16x16 tile of S = F * F^T via V_WMMA_F32_16X16X4_F32,
// then fuses the distance/logit epilogue:
//   lm[i,j] = (i==j) ? 0 : -sqrt(max(0, sq_i + sq_j - 2*S_ij)) / T,  T = 2.
__global__ void rnc_logits_kernel(const float* __restrict__ F,
                                  const float* __restrict__ sq,
                                  float* __restrict__ LM) {
    const int lane = threadIdx.x & 31;
    const int wave = threadIdx.x >> 5;
    const int tile = blockIdx.x * (blockDim.x >> 5) + wave;
    const int ti = tile >> 8;          // tile / TILES
    const int tj = tile & (TILES - 1); // tile % TILES
    const int i0 = ti * 16;
    const int j0 = tj * 16;
    const int r = lane & 15;   // row (A) / col (B) within tile
    const int h = lane >> 4;   // K-half select

    // A fragment: lane r holds F[i0+r][kb + 2h .. +1]  (16x4 f32 layout)
    // B fragment: lane r holds F[j0+r][kb + 2h .. +1]  (B = F^T, 4x16 layout)
    const float* arow = F + (size_t)(i0 + r) * D + 2 * h;
    const float* brow = F + (size_t)(j0 + r) * D + 2 * h;

    v8f c = {};
#pragma unroll
    for (int kk = 0; kk < D / 4; ++kk) {
        v2f a = *(const v2f*)(arow + kk * 4);
        v2f b = *(const v2f*)(brow + kk * 4);
        // 8 args: (neg_a, A, neg_b, B, c_mod, C, reuse_a, reuse_b)
        c = __builtin_amdgcn_wmma_f32_16x16x4_f32(
                false, a, false, b, (short)0, c, false, false);
    }

    // C/D layout: lane n(0..15) col n, VGPR v -> M=v; lanes 16..31 -> M=v+8.
    const float sqj = sq[j0 + r];
#pragma unroll
    for (int v = 0; v < 8; ++v) {
        const int gi = i0 + v + 8 * h;
        const int gj = j0 + r;
        float lm;
        if (gi == gj) {
            lm = 0.f;                       // reference sets diag logits to 0
        } else {
            float d2 = sq[gi] + sqj - 2.f * c[v];
            d2 = fmaxf(d2, 0.f);
            lm = -sqrtf(d2) * 0.5f;         // -dist / T, T = 2
        }
        LM[(size_t)gi * N + gj] = lm;
    }
}

// ----------------------------------------------- per-row denom + loss pass
// One block per row i. Rows of em/dm/lm live in LDS; outer loop broadcasts
// (em[j], dm[j]); each thread accumulates 16 denom values in registers.
__global__ void rnc_loss_kernel(const float* __restrict__ LM,
                                const float* __restrict__ labels,
                                double* __restrict__ acc) {
    __shared__ float s_em[N];
    __shared__ float s_dm[N];
    __shared__ float s_lm[N];
    __shared__ float s_red[BLK];

    const int i = blockIdx.x;
    const int t = threadIdx.x;
    const float li = labels[i];
    const float* lmrow = LM + (size_t)i * N;

    for (int j = t; j < N; j += BLK) {
        const float lm = lmrow[j];
        s_lm[j] = lm;
        s_em[j] = (j == i) ? 0.f : expf(lm);   // em[i]=0 excludes j==i from denom
        s_dm[j] = fabsf(li - labels[j]);
    }
    __syncthreads();

    float dm_loc[KPT];
    float denom[KPT];
#pragma unroll
    for (int kk = 0; kk < KPT; ++kk) {
        dm_loc[kk] = s_dm[t + kk * BLK];
        denom[kk] = 0.f;
    }

    for (int j = 0; j < N; ++j) {
        const float dmj = s_dm[j];   // uniform address -> LDS broadcast
        const float emj = s_em[j];
#pragma unroll
        for (int kk = 0; kk < KPT; ++kk)
            denom[kk] += (dmj >= dm_loc[kk]) ? emj : 0.f;
    }

    float partial = 0.f;
#pragma unroll
    for (int kk = 0; kk < KPT; ++kk) {
        const int k = t + kk * BLK;
        if (k != i)
            partial += s_lm[k] - logf(denom[kk]);  // denom>0: includes j=k term
    }

    // block reduction
    s_red[t] = partial;
    __syncthreads();
    for (int s = BLK / 2; s > 0; s >>= 1) {
        if (t < s) s_red[t] += s_red[t + s];
        __syncthreads();
    }
    if (t == 0)
        atomicAdd(acc, (double)s_red[0]);
}

// ---------------------------------------------------------------- finalize
__global__ void rnc_final_kernel(const double* __restrict__ acc,
                                 float* __restrict__ out) {
    const double scale = -1.0 / ((double)N * (double)(N - 1));
    out[0] = (float)(acc[0] * scale);
}

// ----------------------------------------------------------------- launch
extern "C" void kernel_launch(void* const* d_in, const int* in_sizes, int n_in,
                              void* d_out, int out_size, void* d_ws, size_t ws_size,
                              hipStream_t stream) {
    const float* features = (const float*)d_in[0];   // [4096, 64] fp32
    const float* labels   = (const float*)d_in[1];   // [4096] fp32
    float* out = (float*)d_out;                      // scalar fp32

    char* ws = (char*)d_ws;
    double* acc = (double*)(ws);                 // 8 B accumulator
    float*  sq  = (float*)(ws + 1024);           // 4096 floats (16 KB)
    float*  LM  = (float*)(ws + 17408);          // 4096*4096 floats (64 MB)

    rnc_init_kernel<<<1, 1, 0, stream>>>(acc);

    rnc_sq_kernel<<<N / 256, 256, 0, stream>>>(features, sq);

    // 256x256 = 65536 tiles, 4 waves (128 threads) per block
    rnc_logits_kernel<<<(TILES * TILES) / 4, 128, 0, stream>>>(features, sq, LM);

    rnc_loss_kernel<<<N, BLK, 0, stream>>>(LM, labels, acc);

    rnc_final_kernel<<<1, 1, 0, stream>>>(acc, out);
}